// EmerGNN_8607114461437
// MI455X (gfx1250) — compile-verified
//
#include <hip/hip_runtime.h>
#include <hip/hip_bf16.h>

#define N_ENT 20000
#define BQ    32     // batch
#define D     32     // n_dim
#define LYR   2
#define R     41
#define EVAL_REL 86
#define NEDGE 100000

typedef __attribute__((ext_vector_type(2))) float v2f;
typedef __attribute__((ext_vector_type(8))) float v8f;

// ---------------------------------------------------------------------------
// Prep: ht = [h_emb | t_emb]  (B x 2D), hidden = relu(ht @ relw_w + relw_b),
// rel_w = sigmoid(hidden @ attn_w + attn_b)  -> [L][B][R].
// Also writes h_emb/t_emb into emb[:, 0:64].
// ---------------------------------------------------------------------------
__global__ void prep_kernel(const int* __restrict__ head, const int* __restrict__ tail,
                            const float* __restrict__ ent_kg,
                            const float* __restrict__ relw_w, const float* __restrict__ relw_b,
                            const float* __restrict__ attn_w, const float* __restrict__ attn_b,
                            float* __restrict__ relw_out, float* __restrict__ emb_ws) {
  __shared__ float ht[BQ * 2 * D];        // 2048 floats
  __shared__ float hidden[LYR * BQ * 5];  // 320 floats
  const int t = threadIdx.x;

  for (int i = t; i < BQ * 2 * D; i += blockDim.x) {
    int b = i >> 6, c = i & 63;
    float v = (c < D) ? ent_kg[head[b] * D + c] : ent_kg[tail[b] * D + (c - D)];
    ht[i] = v;
    emb_ws[b * 4 * D + c] = v;  // h_emb at [0,32), t_emb at [32,64)
  }
  __syncthreads();

  for (int i = t; i < LYR * BQ * 5; i += blockDim.x) {
    int l = i / (BQ * 5), rem = i % (BQ * 5), b = rem / 5, j = rem % 5;
    float s = relw_b[l * 5 + j];
    for (int k = 0; k < 2 * D; ++k)
      s += ht[b * 2 * D + k] * relw_w[l * 2 * D * 5 + k * 5 + j];
    hidden[i] = fmaxf(s, 0.f);
  }
  __syncthreads();

  for (int i = t; i < LYR * BQ * R; i += blockDim.x) {
    int l = i / (BQ * R), rem = i % (BQ * R), b = rem / R, r = rem % R;
    float s = attn_b[l * R + r];
    for (int j = 0; j < 5; ++j)
      s += hidden[l * BQ * 5 + b * 5 + j] * attn_w[l * 5 * R + j * R + r];
    relw_out[i] = 1.f / (1.f + __expf(-s));
  }
}

// ---------------------------------------------------------------------------
// Scatter initial embedding: hid[idx[b], b, :] = ent_kg[idx[b], :]
// ---------------------------------------------------------------------------
__global__ void scatter_init_kernel(float* __restrict__ hid, const int* __restrict__ idx,
                                    const float* __restrict__ ent_kg) {
  int t = threadIdx.x;                 // 1024 threads
  int b = t >> 5, d = t & 31;
  int n = idx[b];
  hid[((size_t)n * BQ + b) * D + d] = ent_kg[n * D + d];
}

// ---------------------------------------------------------------------------
// Edge message passing: hid_out[dst] += ew * rel_w[b,etype] * rel_kg[etype,:] * hid_in[src]
// One block per edge, 256 threads -> 4 floats each (B*D = 1024).
// hid fits in L2 (2 x 82MB < 192MB) so atomics resolve in L2.
// ---------------------------------------------------------------------------
__global__ void msg_kernel(const float* __restrict__ hid_in, float* __restrict__ hid_out,
                           const int* __restrict__ src, const int* __restrict__ dst,
                           const int* __restrict__ etype, const float* __restrict__ ew,
                           const float* __restrict__ rel_kg_l, const float* __restrict__ relw_l) {
  const int e = blockIdx.x;
  const int s = src[e], dn = dst[e], et = etype[e];
  const float w = ew[e];
  const int t = threadIdx.x;
  const int b = t >> 3, dq = (t & 7) << 2;

  const float coef = w * relw_l[b * R + et];
  const float4 h  = *reinterpret_cast<const float4*>(hid_in + ((size_t)s * BQ + b) * D + dq);
  const float4 rk = *reinterpret_cast<const float4*>(rel_kg_l + et * D + dq);
  float* o = hid_out + ((size_t)dn * BQ + b) * D + dq;
  atomicAdd(o + 0, coef * rk.x * h.x);
  atomicAdd(o + 1, coef * rk.y * h.y);
  atomicAdd(o + 2, coef * rk.z * h.z);
  atomicAdd(o + 3, coef * rk.w * h.w);
}

// ---------------------------------------------------------------------------
// Batched 32x32 @ 32x32 + bias, ReLU via V_WMMA_F32_16X16X4_F32.
// One wave32 per node; 2x2 output tiles x 8 K-steps = 32 WMMAs/node.
// VGPR layouts per ISA 7.12.2:
//   A (16x4):  lanes 0-15 -> K=0,1 ; lanes 16-31 -> K=2,3 (M = lane&15)
//   B (4x16):  VGPR0 rows K=0 / K=2 by lane half, VGPR1 rows K=1 / K=3 (N = lane&15)
//   C/D 16x16: VGPR i -> M = i (+8 for upper lanes), N = lane&15
// ---------------------------------------------------------------------------
__global__ void linear_wmma_kernel(const float* __restrict__ X_all, float* __restrict__ Y_all,
                                   const float* __restrict__ W, const float* __restrict__ bias) {
  const int wave = threadIdx.x >> 5;
  const int node = blockIdx.x * 8 + wave;
  if (node >= N_ENT) return;                 // wave-uniform: EXEC stays all-ones
  const int lane = threadIdx.x & 31;
  const int half = lane >> 4;
  const int l15  = lane & 15;

  const float* X = X_all + (size_t)node * (BQ * D);
  float*       Y = Y_all + (size_t)node * (BQ * D);

  for (int mt = 0; mt < 2; ++mt) {
    const float* xrow = X + (mt * 16 + l15) * D;
    for (int nt = 0; nt < 2; ++nt) {
      const int col = nt * 16 + l15;
      const float bval = bias[col];
      v8f acc = {bval, bval, bval, bval, bval, bval, bval, bval};
#pragma unroll
      for (int k0 = 0; k0 < D; k0 += 4) {
        const int ka = k0 + half * 2;
        v2f a;  a.x  = xrow[ka];            a.y  = xrow[ka + 1];
        v2f bb; bb.x = W[ka * D + col];     bb.y = W[(ka + 1) * D + col];
        acc = __builtin_amdgcn_wmma_f32_16x16x4_f32(
            false, a, false, bb, (short)0, acc, false, false);
      }
#pragma unroll
      for (int i = 0; i < 8; ++i) {
        const int row = mt * 16 + i + half * 8;
        Y[row * D + col] = fmaxf(acc[i], 0.f);
      }
    }
  }
}

// ---------------------------------------------------------------------------
// Gather: emb[b, sect:sect+32] = hid[idx[b], b, :]
// ---------------------------------------------------------------------------
__global__ void gather_kernel(const float* __restrict__ hid, const int* __restrict__ idx,
                              float* __restrict__ emb_ws, int sect) {
  int t = threadIdx.x;  // 1024 threads
  int b = t >> 5, d = t & 31;
  emb_ws[b * 4 * D + sect + d] = hid[((size_t)idx[b] * BQ + b) * D + d];
}

// ---------------------------------------------------------------------------
// Final: out = emb(32x128) @ Wr_w(128x86) + Wr_b
// ---------------------------------------------------------------------------
__global__ void final_kernel(const float* __restrict__ emb, const float* __restrict__ Wr_w,
                             const float* __restrict__ Wr_b, float* __restrict__ out) {
  int i = blockIdx.x * blockDim.x + threadIdx.x;
  if (i >= BQ * EVAL_REL) return;
  int b = i / EVAL_REL, o = i % EVAL_REL;
  float s = Wr_b[o];
  for (int k = 0; k < 4 * D; ++k)
    s += emb[b * 4 * D + k] * Wr_w[k * EVAL_REL + o];
  out[i] = s;
}

extern "C" void kernel_launch(void* const* d_in, const int* in_sizes, int n_in,
                              void* d_out, int out_size, void* d_ws, size_t ws_size,
                              hipStream_t stream) {
  const int*   head   = (const int*)d_in[0];
  const int*   tail   = (const int*)d_in[1];
  const int*   src    = (const int*)d_in[2];
  const int*   dst    = (const int*)d_in[3];
  const int*   etype  = (const int*)d_in[4];
  const float* ew     = (const float*)d_in[5];
  const float* ent_kg = (const float*)d_in[6];
  const float* rel_kg = (const float*)d_in[7];
  const float* lin_w  = (const float*)d_in[8];
  const float* lin_b  = (const float*)d_in[9];
  const float* relw_w = (const float*)d_in[10];
  const float* relw_b = (const float*)d_in[11];
  const float* attn_w = (const float*)d_in[12];
  const float* attn_b = (const float*)d_in[13];
  const float* Wr_w   = (const float*)d_in[14];
  const float* Wr_b   = (const float*)d_in[15];

  float* ws = (float*)d_ws;
  const size_t NBD = (size_t)N_ENT * BQ * D;     // 20.48M floats
  float* hidA = ws;                              // current state
  float* hidB = ws + NBD;                        // message accumulator
  float* relw = ws + 2 * NBD;                    // [L][B][R]
  float* emb  = relw + LYR * BQ * R;             // [B][4D]

  prep_kernel<<<1, 256, 0, stream>>>(head, tail, ent_kg, relw_w, relw_b,
                                     attn_w, attn_b, relw, emb);

  for (int dir = 0; dir < 2; ++dir) {
    const int* init_idx = (dir == 0) ? head : tail;
    hipMemsetAsync(hidA, 0, NBD * sizeof(float), stream);
    scatter_init_kernel<<<1, 1024, 0, stream>>>(hidA, init_idx, ent_kg);

    for (int l = 0; l < LYR; ++l) {
      hipMemsetAsync(hidB, 0, NBD * sizeof(float), stream);
      msg_kernel<<<NEDGE, 256, 0, stream>>>(hidA, hidB, src, dst, etype, ew,
                                            rel_kg + l * R * D, relw + l * BQ * R);
      linear_wmma_kernel<<<(N_ENT + 7) / 8, 256, 0, stream>>>(hidB, hidA,
                                                              lin_w + l * D * D,
                                                              lin_b + l * D);
    }
    // dir 0 (head->tail): gather at tail -> tail_hid at emb[:,96:128]
    // dir 1 (tail->head): gather at head -> head_hid at emb[:,64:96]
    gather_kernel<<<1, 1024, 0, stream>>>(hidA, (dir == 0) ? tail : head, emb,
                                          (dir == 0) ? 96 : 64);
  }

  final_kernel<<<(BQ * EVAL_REL + 255) / 256, 256, 0, stream>>>(emb, Wr_w, Wr_b,
                                                                (float*)d_out);
}